// SelfAttention3D_28570122453410
// MI455X (gfx1250) — compile-verified
//
#include <hip/hip_runtime.h>

// MI455X (gfx1250) fused 3D self-attention:
//   k1: QKV 1x1x1-conv projections (LDS-cached weights)
//   k2: flash attention with V_WMMA_F32_16X16X4_F32 (8 WMMA / 16x16 tile)
//   k3: Wo projection + bias + gamma*out + x residual
// fp32 end-to-end (exact vs reference up to reassociation), WMMA for both GEMMs.

typedef __attribute__((ext_vector_type(2))) float v2f;
typedef __attribute__((ext_vector_type(8))) float v8f;

#define IN_CH  128
#define INTER  16
#define NVOX   9216          // 16*24*24
#define NTILES (NVOX / 16)   // 576
#define WPB    4             // waves per block in attention kernel

__device__ __forceinline__ v8f wmma4(v2f a, v2f b, v8f c) {
  // D(16x16,f32) = A(16x4,f32) x B(4x16,f32) + C ; args mirror verified f16 pattern
  return __builtin_amdgcn_wmma_f32_16x16x4_f32(false, a, false, b, (short)0, c,
                                               false, false);
}

// exchange with partner half-wave (lane ^ 16) via ds_swizzle SWAPX16
__device__ __forceinline__ float swap16(float x) {
  return __int_as_float(__builtin_amdgcn_ds_swizzle(__float_as_int(x), 0x401F));
}

// ---------------- kernel 1: q/k/v projections ----------------
__global__ __launch_bounds__(256)
void qkv_proj(const float* __restrict__ x,
              const float* __restrict__ Wq, const float* __restrict__ bq,
              const float* __restrict__ Wk, const float* __restrict__ bk,
              const float* __restrict__ Wv, const float* __restrict__ bv,
              float* __restrict__ Qt, float* __restrict__ Kt,
              float* __restrict__ Vt) {
  __shared__ float sW[3 * INTER * IN_CH + 48];
  float* sq = sW;
  float* sk = sW + INTER * IN_CH;
  float* sv = sW + 2 * INTER * IN_CH;
  float* sb = sW + 3 * INTER * IN_CH;
  const int tid = threadIdx.x;
  for (int t = tid; t < INTER * IN_CH; t += 256) {
    sq[t] = Wq[t]; sk[t] = Wk[t]; sv[t] = Wv[t];
  }
  if (tid < 48)
    sb[tid] = (tid < 16) ? bq[tid] : (tid < 32 ? bk[tid - 16] : bv[tid - 32]);
  __syncthreads();

  const int b = blockIdx.y;
  const int n = blockIdx.x * 256 + tid;
  const float* xb = x + (size_t)b * IN_CH * NVOX + n;

  float aq[INTER], ak[INTER], av[INTER];
#pragma unroll
  for (int i = 0; i < INTER; ++i) { aq[i] = sb[i]; ak[i] = sb[16 + i]; av[i] = sb[32 + i]; }

  for (int c = 0; c < IN_CH; ++c) {
    const float xv = xb[(size_t)c * NVOX];
#pragma unroll
    for (int i = 0; i < INTER; ++i) {
      aq[i] = fmaf(sq[i * IN_CH + c], xv, aq[i]);
      ak[i] = fmaf(sk[i * IN_CH + c], xv, ak[i]);
      av[i] = fmaf(sv[i * IN_CH + c], xv, av[i]);
    }
  }
  const size_t row = ((size_t)b * NVOX + n) * INTER;
#pragma unroll
  for (int i = 0; i < INTER; ++i) {
    Qt[row + i] = aq[i]; Kt[row + i] = ak[i]; Vt[row + i] = av[i];
  }
}

// ---------------- kernel 2: flash attention (WMMA f32 16x16x4) ----------------
// One wave owns 16 queries; streams all 576 key tiles with online softmax.
// S^T = K x Q^T  (M=keys, N=queries)  -> softmax along VGPR dim + swap16
// O^T = V^T x P^T (M=head dim, N=queries) -> rescale/normalize is per-lane.
__global__ __launch_bounds__(32 * WPB)
void attn_kernel(const float* __restrict__ Qt, const float* __restrict__ Kt,
                 const float* __restrict__ Vt, float* __restrict__ Ot) {
  const int lane = threadIdx.x & 31;
  const int wid  = threadIdx.x >> 5;
  const int col  = lane & 15;    // query column (and key row / head dim row for A ops)
  const int h    = lane >> 4;    // half-wave select
  const int b    = blockIdx.y;
  const int qtile = blockIdx.x * WPB + wid;   // 0..575

  // Q^T B-operand: lanes hold query=col, VGPR pair holds dims {4j+2h, 4j+2h+1}
  const float* Qb = Qt + ((size_t)b * NVOX + (size_t)qtile * 16) * INTER;
  v2f qb[4];
#pragma unroll
  for (int j = 0; j < 4; ++j)
    qb[j] = *(const v2f*)(Qb + col * INTER + 4 * j + 2 * h);

  v8f o;
#pragma unroll
  for (int r = 0; r < 8; ++r) o[r] = 0.0f;
  float m = -__builtin_inff();
  float l = 0.0f;

  const float* Kb = Kt + (size_t)b * NVOX * INTER;
  const float* Vb = Vt + (size_t)b * NVOX * INTER;

  for (int kt = 0; kt < NTILES; ++kt) {
    const float* Kr = Kb + (size_t)kt * 16 * INTER;
    const float* Vr = Vb + (size_t)kt * 16 * INTER;

    // S^T tile: C-layout rows = keys 8h+r, cols = query col
    v8f s;
#pragma unroll
    for (int r = 0; r < 8; ++r) s[r] = 0.0f;
#pragma unroll
    for (int j = 0; j < 4; ++j) {
      v2f ka = *(const v2f*)(Kr + col * INTER + 4 * j + 2 * h);  // A: M=key=col
      s = wmma4(ka, qb[j], s);
    }

    // ---- online softmax over keys ----
    float tmax = s[0];
#pragma unroll
    for (int r = 1; r < 8; ++r) tmax = fmaxf(tmax, s[r]);
    tmax = fmaxf(tmax, swap16(tmax));
    const float mnew = fmaxf(m, tmax);
    const float resc = __expf(m - mnew);      // first iter: exp(-inf)=0
    float p[8];
    float ps = 0.0f;
#pragma unroll
    for (int r = 0; r < 8; ++r) { p[r] = __expf(s[r] - mnew); ps += p[r]; }
    ps += swap16(ps);
    l = l * resc + ps;
    m = mnew;
#pragma unroll
    for (int r = 0; r < 8; ++r) o[r] *= resc;  // per-lane: O^T cols are queries

    // P^T as B operand: need keys {4j+2h, 4j+2h+1}; local half owns keys 8h+r
    float sh[8];
#pragma unroll
    for (int r = 0; r < 8; ++r) sh[r] = swap16(p[r]);
    v2f pb[4];
    pb[0][0] = h ? sh[2] : p[0];  pb[0][1] = h ? sh[3] : p[1];
    pb[1][0] = h ? sh[6] : p[4];  pb[1][1] = h ? sh[7] : p[5];
    pb[2][0] = h ? p[2]  : sh[0]; pb[2][1] = h ? p[3]  : sh[1];
    pb[3][0] = h ? p[6]  : sh[4]; pb[3][1] = h ? p[7]  : sh[5];

    // V^T A-operand: M=head dim=col, K=keys {4j+2h, 4j+2h+1}
#pragma unroll
    for (int j = 0; j < 4; ++j) {
      v2f va;
      va[0] = Vr[(4 * j + 2 * h)     * INTER + col];
      va[1] = Vr[(4 * j + 2 * h + 1) * INTER + col];
      o = wmma4(va, pb[j], o);
    }
  }

  const float linv = 1.0f / l;
  // Ot layout [b][i][n] : lane col = n offset, VGPR r = i = 8h+r (coalesced rows)
  float* Or = Ot + (size_t)b * INTER * NVOX + (size_t)qtile * 16 + col;
#pragma unroll
  for (int r = 0; r < 8; ++r)
    Or[(size_t)(8 * h + r) * NVOX] = o[r] * linv;
}

// ---------------- kernel 3: output projection + residual ----------------
__global__ __launch_bounds__(256)
void out_proj(const float* __restrict__ Ot, const float* __restrict__ Wo,
              const float* __restrict__ bo, const float* __restrict__ gamma,
              const float* __restrict__ x, float* __restrict__ out) {
  const int n  = blockIdx.x * 256 + threadIdx.x;
  const int c0 = blockIdx.y * 8;
  const int b  = blockIdx.z;
  float ov[INTER];
#pragma unroll
  for (int i = 0; i < INTER; ++i)
    ov[i] = Ot[((size_t)b * INTER + i) * NVOX + n];
  const float g = gamma[0];
#pragma unroll
  for (int cc = 0; cc < 8; ++cc) {
    const int c = c0 + cc;
    const float* w = Wo + c * INTER;
    float acc = bo[c];
#pragma unroll
    for (int i = 0; i < INTER; ++i) acc = fmaf(w[i], ov[i], acc);
    const size_t idx = ((size_t)b * IN_CH + c) * NVOX + n;
    out[idx] = g * acc + x[idx];
  }
}

extern "C" void kernel_launch(void* const* d_in, const int* in_sizes, int n_in,
                              void* d_out, int out_size, void* d_ws, size_t ws_size,
                              hipStream_t stream) {
  const float* x     = (const float*)d_in[0];
  const float* Wq    = (const float*)d_in[1];
  const float* bq    = (const float*)d_in[2];
  const float* Wk    = (const float*)d_in[3];
  const float* bk    = (const float*)d_in[4];
  const float* Wv    = (const float*)d_in[5];
  const float* bv    = (const float*)d_in[6];
  const float* Wo    = (const float*)d_in[7];
  const float* bo    = (const float*)d_in[8];
  const float* gamma = (const float*)d_in[9];
  float* out = (float*)d_out;

  const size_t perArr = (size_t)2 * NVOX * INTER;  // B=2
  float* Qt = (float*)d_ws;
  float* Kt = Qt + perArr;
  float* Vt = Kt + perArr;
  float* Ot = Vt + perArr;                         // total 4.7 MB of d_ws

  qkv_proj<<<dim3(NVOX / 256, 2), 256, 0, stream>>>(x, Wq, bq, Wk, bk, Wv, bv,
                                                    Qt, Kt, Vt);
  attn_kernel<<<dim3(NTILES / WPB, 2), 32 * WPB, 0, stream>>>(Qt, Kt, Vt, Ot);
  out_proj<<<dim3(NVOX / 256, IN_CH / 8, 2), 256, 0, stream>>>(Ot, Wo, bo, gamma,
                                                               x, out);
}